// ScheduleAttention_17575006175252
// MI455X (gfx1250) — compile-verified
//
#include <hip/hip_runtime.h>
#include <hip/hip_bf16.h>

// ---------------------------------------------------------------------------
// Decode-step attention for B=32, S=1, E=2048, H=16, D=128, T=1024 (fp32).
// Memory-bound: ~600MB traffic (KV cache 536MB) vs ~1.34 GFLOP.
//  - GEMMs: V_WMMA_F32_16X16X4_F32, one 16x16 tile per wave.
//  - Attention: KV streamed ONCE with non-temporal loads (protects the
//    67MB of weights resident in the 192MB L2), coalesced 512B rows.
// ---------------------------------------------------------------------------

typedef __attribute__((ext_vector_type(2))) float v2f;
typedef __attribute__((ext_vector_type(4))) float v4f;
typedef __attribute__((ext_vector_type(8))) float v8f;

#define BN 32
#define HH 16
#define DD 128
#define EE 2048
#define TT 1024
#define FF 6144  // 3*H*D

// ---------------------------------------------------------------------------
// Generic C[m,n] = sum_k A[m,k] * W[n,k]   (W row-major [N x K], i.e. B = W^T)
// One 16x16 tile per wave, 4 waves per block covering 4 consecutive N tiles.
// grid = (N/64, M/16), block = 128.
// A fragment (16x4 f32): lane = m + 16*(k>>1), vgpr = k&1  (ISA 7.12.2)
// B fragment symmetric with n in place of m.
// C/D: vgpr r -> row r (lanes 0-15) / row r+8 (lanes 16-31), col = lane&15.
// ---------------------------------------------------------------------------
__global__ __launch_bounds__(128) void wmma_gemm_f32(
    const float* __restrict__ A, const float* __restrict__ W,
    float* __restrict__ C, int K, int lda, int ldb, int ldc)
{
  const int lane = threadIdx.x & 31;
  const int wave = threadIdx.x >> 5;
  const int n0 = (blockIdx.x * 4 + wave) * 16;
  const int m0 = blockIdx.y * 16;
  const int r  = lane & 15;   // m for A-frag, n for B-frag
  const int kh = lane >> 4;   // K half selector

  const float* arow = A + (size_t)(m0 + r) * lda + 2 * kh;
  const float* wrow = W + (size_t)(n0 + r) * ldb + 2 * kh;

  v8f acc = {};
  for (int k0 = 0; k0 < K; k0 += 4) {
    v2f a = { arow[k0], arow[k0 + 1] };
    v2f b = { wrow[k0], wrow[k0 + 1] };
    acc = __builtin_amdgcn_wmma_f32_16x16x4_f32(
        /*neg_a=*/false, a, /*neg_b=*/false, b,
        /*c_mod=*/(short)0, acc, /*reuse_a=*/false, /*reuse_b=*/false);
  }

  float* crow = C + (size_t)(m0 + 8 * kh) * ldc + n0 + r;
#pragma unroll
  for (int i = 0; i < 8; ++i) crow[(size_t)i * ldc] = acc[i];
}

// ---------------------------------------------------------------------------
// Per-(b,h): RMSNorm(q), RMSNorm(k), RoPE(q), RoPE(k), pass-through v.
// grid = 512 (b*16+h), block = 128 (one thread per d).
// proj layout: proj[b*6144 + c*2048 + h*128 + d], c in {0=q,1=k,2=v}.
// ---------------------------------------------------------------------------
__global__ __launch_bounds__(128) void qk_prep(
    const float* __restrict__ proj, const float* __restrict__ qw,
    const float* __restrict__ kw, const int* __restrict__ pos_ids,
    float* __restrict__ qo, float* __restrict__ ko, float* __restrict__ vo)
{
  const int bh = blockIdx.x;
  const int b = bh >> 4, h = bh & 15;
  const int d = threadIdx.x;
  const int lane = d & 31, wave = d >> 5;

  const float* base = proj + (size_t)b * FF + h * DD + d;
  const float qd = base[0];
  const float kd = base[2048];
  const float vd = base[4096];

  // sum of squares across 128 threads (wave shuffle + LDS across 4 waves)
  float sq = qd * qd, sk = kd * kd;
#pragma unroll
  for (int off = 16; off; off >>= 1) {
    sq += __shfl_xor(sq, off, 32);
    sk += __shfl_xor(sk, off, 32);
  }
  __shared__ float rq[4], rk[4], s_rs[2], s_q[128], s_k[128];
  if (lane == 0) { rq[wave] = sq; rk[wave] = sk; }
  __syncthreads();
  if (d == 0) {
    float a = rq[0] + rq[1] + rq[2] + rq[3];
    float c = rk[0] + rk[1] + rk[2] + rk[3];
    s_rs[0] = rsqrtf(a * (1.0f / 128.0f) + 1e-5f);
    s_rs[1] = rsqrtf(c * (1.0f / 128.0f) + 1e-5f);
  }
  __syncthreads();

  const float qn = qd * s_rs[0] * qw[d];
  const float kn = kd * s_rs[1] * kw[d];
  s_q[d] = qn; s_k[d] = kn;
  __syncthreads();

  // RoPE: timescale = 10000^(i/64), i = d mod 64
  const float pos = (float)pos_ids[0];
  const int i = d & 63;
  const float ts = __powf(10000.0f, (float)i * (1.0f / 64.0f));
  const float ang = pos / ts;
  float s, c;
  __sincosf(ang, &s, &c);
  float q_out, k_out;
  if (d < 64) {
    q_out = s_q[d] * c - s_q[d + 64] * s;
    k_out = s_k[d] * c - s_k[d + 64] * s;
  } else {
    q_out = s_q[d] * c + s_q[d - 64] * s;
    k_out = s_k[d] * c + s_k[d - 64] * s;
  }
  qo[(size_t)bh * DD + d] = q_out;
  ko[(size_t)bh * DD + d] = k_out;
  vo[(size_t)bh * DD + d] = vd;
}

// ---------------------------------------------------------------------------
// Decode attention, one block per (b,h). 1025 keys (1024 cache + 1 new).
// Pass 1: one key per wave, float4 coalesced NT row read + shuffle-reduced dot.
// Softmax over 1025 scores in LDS. Pass 2: thread-per-d coalesced NT V stream.
// KV loads are non-temporal: the 536MB stream must not evict the resident
// weight matrices (67MB) from the 192MB L2.
// grid = 512, block = 256 (8 waves).
// ---------------------------------------------------------------------------
__global__ __launch_bounds__(256) void attn_decode(
    const float* __restrict__ q, const float* __restrict__ kc,
    const float* __restrict__ vc, const float* __restrict__ knew,
    const float* __restrict__ vnew, float* __restrict__ out)
{
  const int bh = blockIdx.x;
  const int t = threadIdx.x;
  const int lane = t & 31, wave = t >> 5;

  __shared__ float qs[DD];
  __shared__ float scores[1025];
  __shared__ float red[256];
  __shared__ float part[DD];
  __shared__ float s_scale;

  if (t < DD) qs[t] = q[(size_t)bh * DD + t];
  __syncthreads();

  const float* kbase = kc + (size_t)bh * TT * DD;
  const v4f qv = ((const v4f*)qs)[lane];

  // ---- pass 1: scores[k] = q . k_all[k] ----
  for (int key = wave; key < 1025; key += 8) {
    const float* krow = (key < TT) ? (kbase + (size_t)key * DD)
                                   : (knew + (size_t)bh * DD);
    if (key + 8 < TT)  // hint the stream (global_prefetch_b8)
      __builtin_prefetch(kbase + (size_t)(key + 8) * DD + 4 * lane, 0, 1);
    const v4f kv =
        __builtin_nontemporal_load((const v4f*)(krow + 4 * lane));
    float p = qv.x * kv.x + qv.y * kv.y + qv.z * kv.z + qv.w * kv.w;
#pragma unroll
    for (int off = 16; off; off >>= 1) p += __shfl_xor(p, off, 32);
    if (lane == 0) scores[key] = p;
  }
  __syncthreads();

  // ---- softmax over 1025 scores ----
  float m = -3.4e38f;
  for (int i = t; i < 1025; i += 256) m = fmaxf(m, scores[i]);
  red[t] = m;
  __syncthreads();
  for (int s = 128; s > 0; s >>= 1) {
    if (t < s) red[t] = fmaxf(red[t], red[t + s]);
    __syncthreads();
  }
  m = red[0];
  __syncthreads();

  float sum = 0.0f;
  for (int i = t; i < 1025; i += 256) {
    float e = __expf(scores[i] - m);
    scores[i] = e;
    sum += e;
  }
  red[t] = sum;
  __syncthreads();
  for (int s = 128; s > 0; s >>= 1) {
    if (t < s) red[t] += red[t + s];
    __syncthreads();
  }
  if (t == 0) s_scale = 1.0f / red[0];
  __syncthreads();

  // ---- pass 2: attn[d] = sum_k p[k] * v[k,d], keys split over 2 half-blocks
  const int d = t & 127, half = t >> 7;
  const float* vbase = vc + (size_t)bh * TT * DD + d;
  float acc = 0.0f;
  const int k0 = half ? 513 : 0;
  const int k1 = half ? TT : 513;
  for (int key = k0; key < k1; ++key)
    acc += scores[key] * __builtin_nontemporal_load(vbase + (size_t)key * DD);
  if (half) {
    acc += scores[TT] * vnew[(size_t)bh * DD + d];
    part[d] = acc;
  }
  __syncthreads();
  if (!half) out[(size_t)bh * DD + d] = (acc + part[d]) * s_scale;
}

// ---------------------------------------------------------------------------
extern "C" void kernel_launch(void* const* d_in, const int* in_sizes, int n_in,
                              void* d_out, int out_size, void* d_ws, size_t ws_size,
                              hipStream_t stream) {
  const float* x_t   = (const float*)d_in[0];  // (32,1,2048)
  const float* w_in  = (const float*)d_in[1];  // (6144,2048)
  const float* w_out = (const float*)d_in[2];  // (2048,2048)
  const float* qw    = (const float*)d_in[3];  // (128,)
  const float* kw    = (const float*)d_in[4];  // (128,)
  const float* kc    = (const float*)d_in[5];  // (32,16,1024,128)
  const float* vc    = (const float*)d_in[6];  // (32,16,1024,128)
  const int*   pos   = (const int*)d_in[7];    // (1,)
  float* out = (float*)d_out;                  // (32,1,2048)

  float* ws   = (float*)d_ws;
  float* proj = ws;                       // 32*6144
  float* qb   = proj + (size_t)BN * FF;   // 32*16*128
  float* kb   = qb + (size_t)BN * HH * DD;
  float* vb   = kb + (size_t)BN * HH * DD;
  float* ab   = vb + (size_t)BN * HH * DD;

  // 1) proj = x_t @ w_in^T   (32 x 6144, K=2048)
  wmma_gemm_f32<<<dim3(FF / 64, BN / 16), 128, 0, stream>>>(
      x_t, w_in, proj, EE, EE, EE, FF);

  // 2) rmsnorm + rope -> q, k_new, v_new
  qk_prep<<<BN * HH, DD, 0, stream>>>(proj, qw, kw, pos, qb, kb, vb);

  // 3) attention -> ab (32,16,128), laid out as flat (b, h*128+d)
  attn_decode<<<BN * HH, 256, 0, stream>>>(qb, kc, vc, kb, vb, ab);

  // 4) out = attn_flat @ w_out^T  (32 x 2048, K=2048)
  wmma_gemm_f32<<<dim3(EE / 64, BN / 16), 128, 0, stream>>>(
      ab, w_out, out, EE, EE, EE, EE);
}